// SpotifyGNN_20109036880042
// MI455X (gfx1250) — compile-verified
//
#include <hip/hip_runtime.h>

#define NUM_USERS 200000
#define NUM_ITEMS 100000
#define NN        (NUM_USERS + NUM_ITEMS)   // 300000 nodes
#define EMB       64
#define NEDGES    2000000
#define NLAYERS   3

typedef __attribute__((ext_vector_type(2))) float v2f;
typedef __attribute__((ext_vector_type(8))) float v8f;

// ---- init: x0 = concat(user, item); acc(d_out) = same -----------------------
__global__ void gnn_init(const float* __restrict__ user,
                         const float* __restrict__ item,
                         float* __restrict__ x0,
                         float* __restrict__ acc) {
  size_t idx = (size_t)blockIdx.x * blockDim.x + threadIdx.x;
  const size_t total = (size_t)NN * EMB;
  if (idx >= total) return;
  const size_t uTot = (size_t)NUM_USERS * EMB;
  float v = (idx < uTot) ? user[idx] : item[idx - uTot];
  x0[idx]  = v;
  acc[idx] = v;
}

__global__ void gnn_zero(float* __restrict__ p, size_t n) {
  size_t idx = (size_t)blockIdx.x * blockDim.x + threadIdx.x;
  if (idx < n) p[idx] = 0.0f;
}

// ---- symmetric degree count -------------------------------------------------
__global__ void gnn_degree(const long long* __restrict__ eu,
                           const long long* __restrict__ ei,
                           float* __restrict__ deg) {
  int e = blockIdx.x * blockDim.x + threadIdx.x;
  if (e >= NEDGES) return;
  int u  = (int)eu[e];
  int it = (int)ei[e] + NUM_USERS;
  atomicAdd(&deg[u],  1.0f);
  atomicAdd(&deg[it], 1.0f);
}

// ---- deg -> deg^{-1/2} in place (reference masking semantics) ---------------
__global__ void gnn_dinv(float* __restrict__ deg) {
  int n = blockIdx.x * blockDim.x + threadIdx.x;
  if (n >= NN) return;
  float d = deg[n];
  deg[n] = (d > 0.0f) ? rsqrtf(fmaxf(d, 1.0f)) : 0.0f;
}

// ---- one wave32 per undirected edge: both propagation directions ------------
__global__ void gnn_scatter(const long long* __restrict__ eu,
                            const long long* __restrict__ ei,
                            const float* __restrict__ dinv,
                            const float* __restrict__ x,
                            float* __restrict__ xn) {
  int warp = (int)((blockIdx.x * blockDim.x + threadIdx.x) >> 5);
  int lane = threadIdx.x & 31;
  if (warp >= NEDGES) return;

  // Stream-ahead prefetch of the edge list (lowers to global_prefetch_b8).
  int pf = warp + 4096;
  if (pf >= NEDGES) pf = NEDGES - 1;
  __builtin_prefetch(&eu[pf], 0, 0);
  __builtin_prefetch(&ei[pf], 0, 0);

  int u  = (int)eu[warp];
  int it = (int)ei[warp] + NUM_USERS;
  float norm = dinv[u] * dinv[it];

  // 64 floats per row, 2 per lane, coalesced B64 loads.
  const float2* xu = (const float2*)(x + (size_t)u  * EMB);
  const float2* xi = (const float2*)(x + (size_t)it * EMB);
  float2 a = xu[lane];
  float2 b = xi[lane];

  float* du = xn + (size_t)u  * EMB + 2 * lane;
  float* di = xn + (size_t)it * EMB + 2 * lane;
  atomicAdd(du + 0, b.x * norm);
  atomicAdd(du + 1, b.y * norm);
  atomicAdd(di + 0, a.x * norm);
  atomicAdd(di + 1, a.y * norm);
}

// ---- acc += xn --------------------------------------------------------------
__global__ void gnn_accum(float* __restrict__ acc,
                          const float* __restrict__ xn, size_t n) {
  size_t idx = (size_t)blockIdx.x * blockDim.x + threadIdx.x;
  if (idx < n) acc[idx] += xn[idx];
}

// ---- WMMA epilogue: out *= 0.25 exactly, per 16x16 f32 tile -----------------
// D = sum_j A_j x B_j (+C) with A_j = X[:,4j:4j+4], B_j[k,n] = 0.25*delta(n,4j+k):
// each D[m,n] receives exactly one nonzero product => exact f32 scaling.
// Layouts per CDNA5 ISA 7.12.2:
//   A (16x4 f32, 2 VGPRs): reg r, lanes 0-15: M=lane, K=r; lanes 16-31: M=lane-16, K=r+2
//   B (4x16 f32, 2 VGPRs): reg r, lanes 0-15: K=r, N=lane; lanes 16-31: K=r+2, N=lane-16
//   C/D (16x16 f32, 8 VGPRs): reg r, lanes 0-15: M=r, N=lane; lanes 16-31: M=r+8, N=lane-16
__global__ void gnn_wmma_scale(float* __restrict__ out) {
  int warp = (int)((blockIdx.x * blockDim.x + threadIdx.x) >> 5);
  int lane = threadIdx.x & 31;
  const int ngroups = NN / 16;            // 18750, exact
  if (warp >= ngroups) return;            // wave-uniform: EXEC all-1s below

  int  base  = warp * 16;
  bool low   = (lane < 16);
  int  m0    = lane & 15;
  int  khalf = low ? 0 : 2;

  const float* rowp = out + (size_t)(base + m0) * EMB;

  for (int t = 0; t < 4; ++t) {           // four 16x16 column tiles of the 16x64 block
    int noff = t * 16;
    v8f c = {};
#pragma unroll
    for (int j = 0; j < 4; ++j) {
      v2f a, b;
      a[0] = rowp[noff + 4 * j + khalf + 0];
      a[1] = rowp[noff + 4 * j + khalf + 1];
      b[0] = (m0 == 4 * j + khalf + 0) ? 0.25f : 0.0f;
      b[1] = (m0 == 4 * j + khalf + 1) ? 0.25f : 0.0f;
      c = __builtin_amdgcn_wmma_f32_16x16x4_f32(
          /*neg_a=*/false, a, /*neg_b=*/false, b,
          /*c_mod=*/(short)0, c, /*reuse_a=*/false, /*reuse_b=*/false);
    }
    int mo = low ? 0 : 8;
#pragma unroll
    for (int r = 0; r < 8; ++r) {
      out[(size_t)(base + r + mo) * EMB + noff + m0] = c[r];
    }
  }
}

extern "C" void kernel_launch(void* const* d_in, const int* in_sizes, int n_in,
                              void* d_out, int out_size, void* d_ws, size_t ws_size,
                              hipStream_t stream) {
  const long long* edge = (const long long*)d_in[0];   // [2, NEDGES], int64
  const float*     user = (const float*)d_in[1];       // [NUM_USERS, 64]
  const float*     item = (const float*)d_in[2];       // [NUM_ITEMS, 64]
  const long long* eu = edge;
  const long long* ei = edge + NEDGES;

  float* acc = (float*)d_out;                          // embs_sum accumulator

  const size_t embElems = (size_t)NN * EMB;            // 19.2M floats
  const size_t embBytes = embElems * sizeof(float);    // 76.8 MB
  char* ws = (char*)d_ws;
  float* x0   = (float*)(ws);
  float* x1   = (float*)(ws + embBytes);
  float* dinv = (float*)(ws + 2 * embBytes);           // deg, then deg^-1/2

  const int B = 256;                                   // 8 wave32 waves / block
  const int embBlocks  = (int)((embElems + B - 1) / B);            // 75000
  const int nodeBlocks = (NN + B - 1) / B;                         // 1172
  const int edgeBlocks = (NEDGES + B - 1) / B;                     // 7813
  const int scatBlocks = (int)(((size_t)NEDGES * 32 + B - 1) / B); // 250000
  const int wmmaBlocks = (int)(((size_t)(NN / 16) * 32 + B - 1) / B); // 2344

  // x0 = all_emb; acc = all_emb
  gnn_init<<<embBlocks, B, 0, stream>>>(user, item, x0, acc);

  // degrees -> dinv
  gnn_zero<<<nodeBlocks, B, 0, stream>>>(dinv, (size_t)NN);
  gnn_degree<<<edgeBlocks, B, 0, stream>>>(eu, ei, dinv);
  gnn_dinv<<<nodeBlocks, B, 0, stream>>>(dinv);

  // 3 propagation layers, ping-pong x0 <-> x1, accumulate into acc
  float* x  = x0;
  float* xn = x1;
  for (int layer = 0; layer < NLAYERS; ++layer) {
    gnn_zero<<<embBlocks, B, 0, stream>>>(xn, embElems);
    gnn_scatter<<<scatBlocks, B, 0, stream>>>(eu, ei, dinv, x, xn);
    gnn_accum<<<embBlocks, B, 0, stream>>>(acc, xn, embElems);
    float* tmp = x; x = xn; xn = tmp;
  }

  // final_emb = embs_sum / 4, in place via exact WMMA scaling
  gnn_wmma_scale<<<wmmaBlocks, B, 0, stream>>>(acc);
}